// StructuralProbe_13460427505803
// MI455X (gfx1250) — compile-verified
//
#include <hip/hip_runtime.h>
#include <hip/hip_bf16.h>

typedef __attribute__((ext_vector_type(2))) float v2f;
typedef __attribute__((ext_vector_type(8))) float v8f;

#define BATCH 32
#define LSEQ  256
#define DIN   1024
#define RPROJ 128

// ---------------------------------------------------------------------------
// Kernel 0: Pt (128x1024) = transpose of proj (1024x128), LDS-tiled.
// Lets stage-1 B fragments be contiguous float2 (global_load_b64) per lane.
// ---------------------------------------------------------------------------
__global__ __launch_bounds__(256)
void sp_transpose_proj(const float* __restrict__ P, float* __restrict__ Pt) {
    __shared__ float tile[16][17];                 // +1 pad: no bank conflicts
    const int n0 = blockIdx.x * 16;                // 8 n-tiles
    const int k0 = blockIdx.y * 16;                // 64 k-tiles
    const int tx = threadIdx.x & 15;
    const int ty = threadIdx.x >> 4;
    tile[ty][tx] = P[(size_t)(k0 + ty) * RPROJ + n0 + tx];   // coalesced read
    __syncthreads();
    Pt[(size_t)(n0 + ty) * DIN + k0 + tx] = tile[tx][ty];    // coalesced write
}

// ---------------------------------------------------------------------------
// Stage 1: T[b] = batch[b] (256x1024) @ proj (1024x128), f32 WMMA 16x16x4.
// One wave computes a 16-row x 64-col stripe (4 accumulators); grid has two
// column halves per row tile for 2x occupancy (A slab stays hot in L2).
// A-frag: lane m=L%16 is row M; lanes<16 hold K=k,k+1; lanes>=16 K=k+2,k+3.
// B-frag via Pt: lane m is col N=16t+m -> float2 at Pt[n][k+2kh].
// ---------------------------------------------------------------------------
__global__ __launch_bounds__(32)
void sp_proj_wmma(const float* __restrict__ batch,
                  const float* __restrict__ Pt,
                  float* __restrict__ T) {
    const int b    = blockIdx.z;
    const int row0 = blockIdx.y * 16;
    const int col0 = blockIdx.x * 64;
    const int lane = threadIdx.x;
    const int m    = lane & 15;
    const int kh   = lane >> 4;          // which K pair this lane owns

    const float* __restrict__ arow =
        batch + ((size_t)(b * LSEQ + row0 + m)) * DIN + 2 * kh;
    const float* __restrict__ pt =
        Pt + ((size_t)(col0 + m)) * DIN + 2 * kh;

    v8f acc[4];
#pragma unroll
    for (int t = 0; t < 4; ++t) acc[t] = (v8f){0.f,0.f,0.f,0.f,0.f,0.f,0.f,0.f};

    for (int k = 0; k < DIN; k += 4) {
        v2f a = *(const v2f*)(arow + k);                 // global_load_b64
#pragma unroll
        for (int t = 0; t < 4; ++t) {
            v2f bb = *(const v2f*)(pt + (size_t)(16 * t) * DIN + k);  // b64
            acc[t] = __builtin_amdgcn_wmma_f32_16x16x4_f32(
                false, a, false, bb, (short)0, acc[t], false, false);
        }
    }

    // D layout: lane L (N = L%16), VGPR i -> row M = i + 8*(L>=16)
    float* __restrict__ trow =
        T + ((size_t)(b * LSEQ + row0 + 8 * kh)) * RPROJ + col0 + m;
#pragma unroll
    for (int t = 0; t < 4; ++t) {
#pragma unroll
        for (int i = 0; i < 8; ++i) {
            trow[(size_t)i * RPROJ + 16 * t] = acc[t][i];
        }
    }
}

// ---------------------------------------------------------------------------
// Stage 2: G = T Tt per batch via WMMA; fused epilogue emits
//   out[i][j] = |t_i|^2 + |t_j|^2 - 2*G[i][j].
// One wave computes a 16(i) x 64(j) tile (4 accumulators). Row norms fall out
// of the same float2 fragments (per-lane sum of squares, halves combined via
// shfl_xor 16; i-row norm gathered per accumulator row with ds_bpermute).
// Output is write-once -> nontemporal stores.
// ---------------------------------------------------------------------------
__global__ __launch_bounds__(32)
void sp_dist_wmma(const float* __restrict__ T,
                  float* __restrict__ out) {
    const int b    = blockIdx.z;
    const int i0   = blockIdx.y * 16;
    const int j0   = blockIdx.x * 64;
    const int lane = threadIdx.x;
    const int m    = lane & 15;
    const int kh   = lane >> 4;

    const float* __restrict__ ti =
        T + ((size_t)(b * LSEQ + i0 + m)) * RPROJ + 2 * kh;
    const float* __restrict__ tj0 =
        T + ((size_t)(b * LSEQ + j0 + m)) * RPROJ + 2 * kh;

    v8f acc[4];
#pragma unroll
    for (int t = 0; t < 4; ++t) acc[t] = (v8f){0.f,0.f,0.f,0.f,0.f,0.f,0.f,0.f};
    float sa = 0.f;
    float sb[4] = {0.f, 0.f, 0.f, 0.f};

    for (int k = 0; k < RPROJ; k += 4) {
        v2f a = *(const v2f*)(ti + k);
        sa += a.x * a.x + a.y * a.y;
#pragma unroll
        for (int t = 0; t < 4; ++t) {
            v2f bb = *(const v2f*)(tj0 + (size_t)(16 * t) * RPROJ + k);
            sb[t] += bb.x * bb.x + bb.y * bb.y;
            acc[t] = __builtin_amdgcn_wmma_f32_16x16x4_f32(
                false, a, false, bb, (short)0, acc[t], false, false);
        }
    }

    // lane L holds |t_{i0 + L%16}|^2 after combining the two K-half lanes
    float ni = sa + __shfl_xor(sa, 16, 32);
    float nj[4];
#pragma unroll
    for (int t = 0; t < 4; ++t) nj[t] = sb[t] + __shfl_xor(sb[t], 16, 32);

    float* __restrict__ orow =
        out + ((size_t)(b * LSEQ + i0 + 8 * kh)) * LSEQ + j0 + m;
#pragma unroll
    for (int i = 0; i < 8; ++i) {
        // norm of output row M = i + 8*kh lives in lane (i + 8*kh) of ni
        float nrmi = __shfl(ni, i + 8 * kh, 32);
#pragma unroll
        for (int t = 0; t < 4; ++t) {
            float val = nrmi + nj[t] - 2.0f * acc[t][i];
            __builtin_nontemporal_store(val, &orow[(size_t)i * LSEQ + 16 * t]);
        }
    }
}

// ---------------------------------------------------------------------------
// Launch: d_in[0] = batch (32*256*1024 f32), d_in[1] = proj (1024*128 f32),
// d_out = 32*256*256 f32. d_ws: [0, 4MiB) = T, [4MiB, 4.5MiB) = proj^T.
// ---------------------------------------------------------------------------
extern "C" void kernel_launch(void* const* d_in, const int* in_sizes, int n_in,
                              void* d_out, int out_size, void* d_ws, size_t ws_size,
                              hipStream_t stream) {
    const float* batch = (const float*)d_in[0];
    const float* proj  = (const float*)d_in[1];
    float*       outp  = (float*)d_out;
    float*       T     = (float*)d_ws;                       // 32*256*128 f32
    float*       Pt    = T + (size_t)BATCH * LSEQ * RPROJ;   // 128*1024  f32

    (void)in_sizes; (void)n_in; (void)out_size; (void)ws_size;

    sp_transpose_proj<<<dim3(RPROJ / 16, DIN / 16, 1), dim3(256, 1, 1), 0, stream>>>(proj, Pt);

    dim3 g1(RPROJ / 64, LSEQ / 16, BATCH);   // 2 col-halves x 16 row-tiles x 32
    sp_proj_wmma<<<g1, dim3(32, 1, 1), 0, stream>>>(batch, Pt, T);

    dim3 g2(LSEQ / 64, LSEQ / 16, BATCH);    // 4 j-groups x 16 i-tiles x 32
    sp_dist_wmma<<<g2, dim3(32, 1, 1), 0, stream>>>(T, outp);
}